// LPPLLayer_84808424227339
// MI455X (gfx1250) — compile-verified
//
#include <hip/hip_runtime.h>
#include <math.h>

typedef __attribute__((ext_vector_type(2))) float v2f;
typedef __attribute__((ext_vector_type(8))) float v8f;

#define NSUM 16      // 13 sums used, padded to 16 for the WMMA tile
#define TPB 256
#define LSTRIDE 17   // LDS row stride (pad to dodge bank conflicts)

__device__ __forceinline__ void lppl_features(float t, float tcf, float m, float w,
                                              float& p, float& g, float& h) {
    float dt = tcf - t;                 // strictly positive (tc > N-1)
    float ln = __logf(dt);              // v_log_f32 * ln2
    p = __expf(m * ln);                 // dt^m = exp(m*ln dt) -> v_exp_f32
    float ang = w * ln;                 // w * ln(dt)
    float s, c;
    __sincosf(ang, &s, &c);             // v_sin_f32 / v_cos_f32
    g = p * c;
    h = p * s;
}

// ---------------------------------------------------------------------------
// Pass 1: per-lane accumulation of the 13 Gram sums, then block reduction of
// the 256x16 partial matrix via V_WMMA_F32_16X16X4_F32 with all-ones A:
// D[i][j] = sum_k B[k][j]  (K-order invariant, only needs N = lane&15).
// ---------------------------------------------------------------------------
__global__ void lppl_reduce(const float* __restrict__ x,
                            const float* __restrict__ kern,
                            float* __restrict__ partials, int T, int nblocks) {
    __shared__ float lds[TPB * LSTRIDE];

    float tcf = kern[0] * 5.0f + (float)T;
    float m   = kern[1];
    float w   = kern[2] * 10.0f;

    float sf=0.f, sg=0.f, sh=0.f, sy=0.f;
    float ff=0.f, fg=0.f, fh=0.f, gg=0.f, gh=0.f, hh=0.f;
    float yf=0.f, yg=0.f, yh=0.f;

    int gid  = blockIdx.x * TPB + threadIdx.x;
    int gsz  = nblocks * TPB;
    int nvec = T >> 2;
    const float4* __restrict__ x4 = (const float4*)x;

    for (int v = gid; v < nvec; v += gsz) {
        if (v + gsz < nvec) __builtin_prefetch(&x4[v + gsz], 0, 0);  // global_prefetch_b8
        float4 xv = x4[v];
        float base = (float)(v * 4);   // exact: < 2^24
        #pragma unroll
        for (int k = 0; k < 4; ++k) {
            float p, g, h;
            lppl_features(base + (float)k, tcf, m, w, p, g, h);
            float y = (&xv.x)[k];
            sf += p;  sg += g;  sh += h;  sy += y;
            ff = fmaf(p, p, ff);  fg = fmaf(p, g, fg);  fh = fmaf(p, h, fh);
            gg = fmaf(g, g, gg);  gh = fmaf(g, h, gh);  hh = fmaf(h, h, hh);
            yf = fmaf(y, p, yf);  yg = fmaf(y, g, yg);  yh = fmaf(y, h, yh);
        }
    }
    // Tail (T not a multiple of 4) — distributed over the grid's first lanes.
    int tail0 = nvec * 4;
    if (gid < T - tail0) {
        int i = tail0 + gid;
        float p, g, h;
        lppl_features((float)i, tcf, m, w, p, g, h);
        float y = x[i];
        sf += p;  sg += g;  sh += h;  sy += y;
        ff = fmaf(p, p, ff);  fg = fmaf(p, g, fg);  fh = fmaf(p, h, fh);
        gg = fmaf(g, g, gg);  gh = fmaf(g, h, gh);  hh = fmaf(h, h, hh);
        yf = fmaf(y, p, yf);  yg = fmaf(y, g, yg);  yh = fmaf(y, h, yh);
    }

    float* row = &lds[threadIdx.x * LSTRIDE];
    row[0]=sf;  row[1]=sg;  row[2]=sh;  row[3]=sy;
    row[4]=ff;  row[5]=fg;  row[6]=fh;  row[7]=gg; row[8]=gh; row[9]=hh;
    row[10]=yf; row[11]=yg; row[12]=yh; row[13]=0.f; row[14]=0.f; row[15]=0.f;
    __syncthreads();

    if (threadIdx.x < 32) {            // wave 0 only: EXEC all-ones within the wave
        int j    = threadIdx.x & 15;
        int half = threadIdx.x >> 4;
        v2f a; a[0] = 1.0f; a[1] = 1.0f;   // all-ones A => D rows = column sums of B
        v8f c = {};
        for (int r = 0; r < TPB; r += 4) {
            v2f b;
            b[0] = lds[(r + half * 2 + 0) * LSTRIDE + j];
            b[1] = lds[(r + half * 2 + 1) * LSTRIDE + j];
            c = __builtin_amdgcn_wmma_f32_16x16x4_f32(false, a, false, b,
                                                      (short)0, c, false, false);
        }
        if (threadIdx.x < 16)           // D[0][j] lives in VGPR0, lanes 0-15
            partials[blockIdx.x * NSUM + threadIdx.x] = c[0];
    }
}

// ---------------------------------------------------------------------------
// Pass 2: one wave reduces the nblocks x 16 partial matrix with the same
// WMMA trick, then thread 0 solves the 4x4 normal equations in double.
// ---------------------------------------------------------------------------
__global__ void lppl_solve(const float* __restrict__ partials,
                           float* __restrict__ coef, int nblocks, int T) {
    __shared__ float sums[16];
    int j    = threadIdx.x & 15;
    int half = threadIdx.x >> 4;
    v2f a; a[0] = 1.0f; a[1] = 1.0f;
    v8f c = {};
    for (int r = 0; r < nblocks; r += 4) {
        v2f b;
        b[0] = partials[(r + half * 2 + 0) * NSUM + j];
        b[1] = partials[(r + half * 2 + 1) * NSUM + j];
        c = __builtin_amdgcn_wmma_f32_16x16x4_f32(false, a, false, b,
                                                  (short)0, c, false, false);
    }
    if (threadIdx.x < 16) sums[threadIdx.x] = c[0];
    __syncthreads();

    if (threadIdx.x == 0) {
        double sf=sums[0], sg=sums[1], sh=sums[2], sy=sums[3];
        double ff=sums[4], fg=sums[5], fh=sums[6];
        double gg=sums[7], gh=sums[8], hh=sums[9];
        double yf=sums[10], yg=sums[11], yh=sums[12];
        double N = (double)T;
        double A[4][5] = {
            { N,  sf, sg, sh, sy },
            { sf, ff, fg, fh, yf },
            { sg, fg, gg, gh, yg },
            { sh, fh, gh, hh, yh },
        };
        // Gauss-Jordan with partial pivoting (double precision)
        for (int col = 0; col < 4; ++col) {
            int piv = col;
            double best = fabs(A[col][col]);
            for (int r = col + 1; r < 4; ++r) {
                double v = fabs(A[r][col]);
                if (v > best) { best = v; piv = r; }
            }
            if (piv != col)
                for (int k = 0; k < 5; ++k) {
                    double t = A[col][k]; A[col][k] = A[piv][k]; A[piv][k] = t;
                }
            double inv = 1.0 / A[col][col];
            for (int r = 0; r < 4; ++r) {
                if (r == col) continue;
                double f = A[r][col] * inv;
                for (int k = col; k < 5; ++k) A[r][k] -= f * A[col][k];
            }
        }
        for (int r = 0; r < 4; ++r)
            coef[r] = (float)(A[r][4] / A[r][r]);
    }
}

// ---------------------------------------------------------------------------
// Pass 3: elementwise model evaluation, float4-vectorized stores.
// ---------------------------------------------------------------------------
__global__ void lppl_apply(const float* __restrict__ kern,
                           const float* __restrict__ coef,
                           float* __restrict__ out, int T) {
    float tcf = kern[0] * 5.0f + (float)T;
    float m   = kern[1];
    float w   = kern[2] * 10.0f;
    float ca = coef[0], cb = coef[1], cc = coef[2], cd = coef[3];

    int gid = blockIdx.x * blockDim.x + threadIdx.x;
    int i0  = gid * 4;
    if (i0 + 3 < T) {
        float4 o;
        #pragma unroll
        for (int k = 0; k < 4; ++k) {
            float p, g, h;
            lppl_features((float)(i0 + k), tcf, m, w, p, g, h);
            (&o.x)[k] = fmaf(cd, h, fmaf(cc, g, fmaf(cb, p, ca)));
        }
        *(float4*)(out + i0) = o;
    } else {
        for (int i = i0; i < T; ++i) {
            float p, g, h;
            lppl_features((float)i, tcf, m, w, p, g, h);
            out[i] = fmaf(cd, h, fmaf(cc, g, fmaf(cb, p, ca)));
        }
    }
}

extern "C" void kernel_launch(void* const* d_in, const int* in_sizes, int n_in,
                              void* d_out, int out_size, void* d_ws, size_t ws_size,
                              hipStream_t stream) {
    const float* x    = (const float*)d_in[0];
    const float* kern = (const float*)d_in[1];
    float* out = (float*)d_out;
    int T = in_sizes[0];

    float* ws       = (float*)d_ws;
    float* coef     = ws;        // 4 floats (first 16 reserved)
    float* partials = ws + 16;   // nblocks * 16 floats

    long maxb = ((long)(ws_size / sizeof(float)) - 16) / NSUM;
    int nblocks = 4096;
    if (maxb < (long)nblocks) nblocks = (int)maxb;
    nblocks &= ~3;               // multiple of 4 for the WMMA K-tiling
    if (nblocks < 4) nblocks = 4;

    lppl_reduce<<<nblocks, TPB, 0, stream>>>(x, kern, partials, T, nblocks);
    lppl_solve<<<1, 32, 0, stream>>>(partials, coef, nblocks, T);

    int apply_blocks = (((T + 3) / 4) + TPB - 1) / TPB;
    lppl_apply<<<apply_blocks, TPB, 0, stream>>>(kern, coef, out, T);
}